// MotionSparseMoeBlock_22814866276628
// MI455X (gfx1250) — compile-verified
//
#include <hip/hip_runtime.h>
#include <hip/hip_bf16.h>
#include <math.h>

// Problem dims (fixed by the reference).
#define Tn 4096   // tokens = B*S
#define Dn 1024   // embed dim
#define En 8      // experts
#define Fn 4096   // ffn dim
// GEMM tiling: 256 threads = 8 wave32 waves, each wave -> 32x32 C tile.
#define BM 64
#define BN 128
#define KT 16
#define SA 20     // LDS A row stride (floats): 16B-aligned float4, low conflicts
#define SBR 144   // LDS B row stride (floats): 128 data + 16 pad (TDM pad) ->
                  // k and k+2 rows land 32 banks apart for the two lane halves

typedef float v2f __attribute__((ext_vector_type(2)));
typedef float v8f __attribute__((ext_vector_type(8)));
typedef unsigned int v4u __attribute__((ext_vector_type(4)));
typedef int v4i __attribute__((ext_vector_type(4)));
typedef int v8i __attribute__((ext_vector_type(8)));

__device__ __forceinline__ int imin(int a, int b) { return a < b ? a : b; }

// ---------------------------------------------------------------------------
// Tensor Data Mover: DMA a 2D tile (rows x tile_w f32, row pitch = stride
// elements) from global memory into LDS at byte offset lds_off, inserting
// pad_amt+1 DWORDs of padding every (1<<(pad_int+1)) DWORDs (D# section 8.3/8.4).
// Issued by one wave; completion tracked with TENSORcnt.
// ---------------------------------------------------------------------------
__device__ __forceinline__ void tdm_load_tile_f32(const void* gptr,
                                                  unsigned lds_off, int tile_w,
                                                  int rows, int stride,
                                                  unsigned pad_int,
                                                  unsigned pad_amt) {
  unsigned long long ga = (unsigned long long)gptr;
  v4u g0;
  g0[0] = 1u;                                   // count=1, user descriptor
  g0[1] = lds_off;                              // lds_addr (bytes)
  g0[2] = (unsigned)(ga & 0xffffffffu);         // global_addr[31:0]
  g0[3] = (unsigned)((ga >> 32) & 0x1ffffffu)   // global_addr[56:32]
          | (2u << 30);                         // type=2 ("image")
  v8i g1;
  g1[0] = (int)((2u << 16)                      // data_size = 4 bytes
                | (1u << 20)                    // pad_enable
                | (pad_int << 22) | (pad_amt << 25));
  g1[1] = (int)((unsigned)tile_w << 16);        // tensor_dim0 = tile_w
  g1[2] = (int)((unsigned)rows << 16);          // tensor_dim1 = rows
  g1[3] = (int)((unsigned)tile_w << 16);        // tile_dim0 = tile_w
  g1[4] = rows;                                 // tile_dim1 = rows, tile_dim2=0
  g1[5] = stride;                               // tensor_dim0_stride (elems)
  g1[6] = 0;
  g1[7] = 0;
  v4i z4 = {0, 0, 0, 0};                        // groups 2/3: unused (2D tile)
  v8i z8 = {0, 0, 0, 0, 0, 0, 0, 0};
  __builtin_amdgcn_tensor_load_to_lds(g0, g1, z4, z4, z8, 0);
}

// ---------------------------------------------------------------------------
// Router: one wave per token. gate/noise logits via strided GEMV + wave
// butterfly reduction, then softplus/noisy logits, top-2 + renorm (== softmax
// over the top-2 logits), atomic scatter into per-expert (token, weight) lists.
// ---------------------------------------------------------------------------
__global__ __launch_bounds__(256) void moe_router(
    const float* __restrict__ x, const float* __restrict__ noise,
    const float* __restrict__ Wg, const float* __restrict__ bg,
    const float* __restrict__ Wn, const float* __restrict__ bn,
    float* __restrict__ noisy_out, float* __restrict__ gate_out,
    int* __restrict__ cnt, int* __restrict__ idx, float* __restrict__ cw) {
  const int lane = threadIdx.x & 31;
  const int t = blockIdx.x * 8 + (threadIdx.x >> 5);
  if (t >= Tn) return;

  float g[En] = {};
  float q[En] = {};
  const float* xr = x + (size_t)t * Dn;
  for (int i = lane; i < Dn; i += 32) {
    float xv = xr[i];
    const float4* wg4 = (const float4*)(Wg + (size_t)i * En);
    const float4* wn4 = (const float4*)(Wn + (size_t)i * En);
    float4 a0 = wg4[0], a1 = wg4[1];
    float4 c0 = wn4[0], c1 = wn4[1];
    g[0] = fmaf(xv, a0.x, g[0]); g[1] = fmaf(xv, a0.y, g[1]);
    g[2] = fmaf(xv, a0.z, g[2]); g[3] = fmaf(xv, a0.w, g[3]);
    g[4] = fmaf(xv, a1.x, g[4]); g[5] = fmaf(xv, a1.y, g[5]);
    g[6] = fmaf(xv, a1.z, g[6]); g[7] = fmaf(xv, a1.w, g[7]);
    q[0] = fmaf(xv, c0.x, q[0]); q[1] = fmaf(xv, c0.y, q[1]);
    q[2] = fmaf(xv, c0.z, q[2]); q[3] = fmaf(xv, c0.w, q[3]);
    q[4] = fmaf(xv, c1.x, q[4]); q[5] = fmaf(xv, c1.y, q[5]);
    q[6] = fmaf(xv, c1.z, q[6]); q[7] = fmaf(xv, c1.w, q[7]);
  }
#pragma unroll
  for (int off = 16; off; off >>= 1) {
#pragma unroll
    for (int e = 0; e < En; ++e) {
      g[e] += __shfl_xor(g[e], off, 32);
      q[e] += __shfl_xor(q[e], off, 32);
    }
  }
  if (lane == 0) {
    float nv[En];
#pragma unroll
    for (int e = 0; e < En; ++e) {
      float gate = g[e] + bg[e];
      float nl = q[e] + bn[e];
      float sp = fmaxf(nl, 0.0f) + log1pf(expf(-fabsf(nl)));  // softplus
      float z = gate + noise[(size_t)t * En + e] * sp;
      nv[e] = z;
      gate_out[(size_t)t * En + e] = gate;
      noisy_out[(size_t)t * En + e] = z;
    }
    int i1 = 0;
#pragma unroll
    for (int e = 1; e < En; ++e)
      if (nv[e] > nv[i1]) i1 = e;
    int i2 = (i1 == 0) ? 1 : 0;
#pragma unroll
    for (int e = 0; e < En; ++e)
      if (e != i1 && nv[e] > nv[i2]) i2 = e;
    float w1 = 1.0f / (1.0f + expf(nv[i2] - nv[i1]));
    float w2 = 1.0f - w1;
    int p1 = atomicAdd(&cnt[i1], 1);
    idx[(size_t)i1 * Tn + p1] = t;
    cw[(size_t)i1 * Tn + p1] = w1;
    int p2 = atomicAdd(&cnt[i2], 1);
    idx[(size_t)i2 * Tn + p2] = t;
    cw[(size_t)i2 * Tn + p2] = w2;
  }
}

// Exclusive prefix over the 8 expert counts (total is exactly 2*Tn).
__global__ void moe_offsets(const int* __restrict__ cnt, int* __restrict__ offs) {
  if (threadIdx.x == 0) {
    int s = 0;
#pragma unroll
    for (int e = 0; e < En; ++e) {
      offs[e] = s;
      s += cnt[e];
    }
    offs[En] = s;
  }
}

// ---------------------------------------------------------------------------
// FC1: h[hb+m, f] = GELU( sum_d x[tok(m), d] * W1[e, d, f] + b1[e, f] )
// A rows gathered via token lists (manual double-buffered path); W1 tile DMA'd
// by the TDM (wave 0) into padded row-major LDS. One barrier per K-chunk.
// ---------------------------------------------------------------------------
__global__ __launch_bounds__(256) void moe_fc1(
    const float* __restrict__ x, const float* __restrict__ W1,
    const float* __restrict__ b1, const int* __restrict__ cnt,
    const int* __restrict__ offs, const int* __restrict__ idx,
    float* __restrict__ h) {
  const int e = blockIdx.z;
  const int count = cnt[e];
  const int m0 = blockIdx.x * BM;
  if (m0 >= count) return;  // uniform per block -> EXEC all-ones for WMMA
  const int f0 = blockIdx.y * BN;
  const int hb = offs[e];

  __shared__ float As[2][BM * SA];
  __shared__ float Bs[2][KT * SBR];

  const int tid = threadIdx.x;
  const int lane = tid & 31;
  const int wv = tid >> 5;
  const int wm = wv & 1;   // 0..1 -> 32-row slice
  const int wn = wv >> 1;  // 0..3 -> 32-col slice

  v8f acc[2][2] = {};

  // A loads: 64x16 floats, one float4 per thread
  const int a_m = tid >> 2;
  const int a_k = (tid & 3) * 4;
  const int tokA = idx[(size_t)e * Tn + imin(m0 + a_m, count - 1)];
  const float* xrow = x + (size_t)tokA * Dn;
  const float* Wbase = W1 + (size_t)e * Dn * Fn + f0;

  const unsigned bs_base = (unsigned)(unsigned long long)(void*)&Bs[0][0];
  const unsigned bs_sz = KT * SBR * 4u;

  const int ml = lane & 15;
  const int kh = (lane >> 4) * 2;
  const int nl0 = wn * 32 + ml;

  // Prologue: stage chunk 0 into buffer 0.
  if (wv == 0)
    tdm_load_tile_f32(Wbase, bs_base, BN, KT, Fn, /*128 DW*/ 6, /*16 DW*/ 15);
  {
    float4 av = *(const float4*)(xrow + a_k);
    *(float4*)(&As[0][a_m * SA + a_k]) = av;
  }

  const int NCH = Dn / KT;
  for (int i = 0; i < NCH; ++i) {
    const int buf = i & 1;
    __builtin_amdgcn_s_wait_tensorcnt(0);  // TDM for buf complete (wave 0)
    __syncthreads();                       // publish buf; prior reads retired
    if (i + 1 < NCH) {                     // prefetch chunk i+1 into buf^1
      const int kn = (i + 1) * KT;
      if (wv == 0)
        tdm_load_tile_f32(Wbase + (size_t)kn * Fn, bs_base + (buf ^ 1) * bs_sz,
                          BN, KT, Fn, 6, 15);
      float4 av = *(const float4*)(xrow + kn + a_k);
      *(float4*)(&As[buf ^ 1][a_m * SA + a_k]) = av;
    }
#pragma unroll
    for (int ks = 0; ks < 4; ++ks) {
      v2f a[2], b[2];
#pragma unroll
      for (int mt = 0; mt < 2; ++mt) {
        const float* p = &As[buf][(wm * 32 + mt * 16 + ml) * SA + ks * 4 + kh];
        a[mt][0] = p[0];
        a[mt][1] = p[1];
      }
#pragma unroll
      for (int nt = 0; nt < 2; ++nt) {
        const float* p = &Bs[buf][(ks * 4 + kh) * SBR + nl0 + nt * 16];
        b[nt][0] = p[0];
        b[nt][1] = p[SBR];
      }
#pragma unroll
      for (int mt = 0; mt < 2; ++mt)
#pragma unroll
        for (int nt = 0; nt < 2; ++nt)
          acc[mt][nt] = __builtin_amdgcn_wmma_f32_16x16x4_f32(
              false, a[mt], false, b[nt], (short)0, acc[mt][nt], false, false);
    }
  }

  const int rh = (lane >> 4) * 8;
#pragma unroll
  for (int mt = 0; mt < 2; ++mt) {
#pragma unroll
    for (int nt = 0; nt < 2; ++nt) {
      const int f = f0 + wn * 32 + nt * 16 + ml;
      const float bias = b1[(size_t)e * Fn + f];
#pragma unroll
      for (int r = 0; r < 8; ++r) {
        const int m = m0 + wm * 32 + mt * 16 + rh + r;
        if (m < count) {
          float v = acc[mt][nt][r] + bias;
          // exact GELU: 0.5*v*(1+erf(v/sqrt(2)))
          float gl = 0.5f * v * (1.0f + erff(v * 0.70710678118654752f));
          h[(size_t)(hb + m) * Fn + f] = gl;
        }
      }
    }
  }
}

// ---------------------------------------------------------------------------
// FC2: out[tok(m), n] += w(m) * ( sum_f h[hb+m, f] * W2[e, f, n] + b2[e, n] )
// Both the A tile (contiguous h rows, pitch F) and the W2 tile are DMA'd by
// the TDM; the inner loop has zero per-thread global loads. Exactly two fp32
// adds land on each output element (fp add commutative -> deterministic).
// ---------------------------------------------------------------------------
__global__ __launch_bounds__(256) void moe_fc2(
    const float* __restrict__ h, const float* __restrict__ W2,
    const float* __restrict__ b2, const int* __restrict__ cnt,
    const int* __restrict__ offs, const int* __restrict__ idx,
    const float* __restrict__ cw, float* __restrict__ out) {
  const int e = blockIdx.z;
  const int count = cnt[e];
  const int m0 = blockIdx.x * BM;
  if (m0 >= count) return;
  const int n0 = blockIdx.y * BN;
  const int hb = offs[e];
  const int rowsA = imin(BM, count - m0);  // clamp: never read past end of h

  __shared__ float As[2][BM * SA];
  __shared__ float Bs[2][KT * SBR];

  const int tid = threadIdx.x;
  const int lane = tid & 31;
  const int wv = tid >> 5;
  const int wm = wv & 1;
  const int wn = wv >> 1;

  v8f acc[2][2] = {};

  const float* Abase = h + (size_t)(hb + m0) * Fn;
  const float* Wbase = W2 + (size_t)e * Fn * Dn + n0;

  const unsigned as_base = (unsigned)(unsigned long long)(void*)&As[0][0];
  const unsigned bs_base = (unsigned)(unsigned long long)(void*)&Bs[0][0];
  const unsigned as_sz = BM * SA * 4u;
  const unsigned bs_sz = KT * SBR * 4u;

  const int ml = lane & 15;
  const int kh = (lane >> 4) * 2;
  const int nl0 = wn * 32 + ml;

  // Prologue: stage chunk 0 (A tile: rowsA x 16 pitch Fn; B tile: 16 x 128).
  if (wv == 0) {
    tdm_load_tile_f32(Abase, as_base, KT, rowsA, Fn, /*16 DW*/ 3, /*4 DW*/ 3);
    tdm_load_tile_f32(Wbase, bs_base, BN, KT, Dn, 6, 15);
  }

  const int NCH = Fn / KT;
  for (int i = 0; i < NCH; ++i) {
    const int buf = i & 1;
    __builtin_amdgcn_s_wait_tensorcnt(0);  // both TDMs for buf complete
    __syncthreads();
    if (i + 1 < NCH) {
      const int kn = (i + 1) * KT;
      if (wv == 0) {
        tdm_load_tile_f32(Abase + kn, as_base + (buf ^ 1) * as_sz, KT, rowsA,
                          Fn, 3, 3);
        tdm_load_tile_f32(Wbase + (size_t)kn * Dn, bs_base + (buf ^ 1) * bs_sz,
                          BN, KT, Dn, 6, 15);
      }
    }
#pragma unroll
    for (int ks = 0; ks < 4; ++ks) {
      v2f a[2], b[2];
#pragma unroll
      for (int mt = 0; mt < 2; ++mt) {
        const float* p = &As[buf][(wm * 32 + mt * 16 + ml) * SA + ks * 4 + kh];
        a[mt][0] = p[0];
        a[mt][1] = p[1];
      }
#pragma unroll
      for (int nt = 0; nt < 2; ++nt) {
        const float* p = &Bs[buf][(ks * 4 + kh) * SBR + nl0 + nt * 16];
        b[nt][0] = p[0];
        b[nt][1] = p[SBR];
      }
#pragma unroll
      for (int mt = 0; mt < 2; ++mt)
#pragma unroll
        for (int nt = 0; nt < 2; ++nt)
          acc[mt][nt] = __builtin_amdgcn_wmma_f32_16x16x4_f32(
              false, a[mt], false, b[nt], (short)0, acc[mt][nt], false, false);
    }
  }

  const int rh = (lane >> 4) * 8;
#pragma unroll
  for (int mt = 0; mt < 2; ++mt) {
#pragma unroll
    for (int nt = 0; nt < 2; ++nt) {
      const int n = n0 + wn * 32 + nt * 16 + ml;
      const float bias = b2[(size_t)e * Dn + n];
#pragma unroll
      for (int r = 0; r < 8; ++r) {
        const int m = m0 + wm * 32 + mt * 16 + rh + r;
        if (m < count) {
          const int tok = idx[(size_t)e * Tn + m];
          const float w = cw[(size_t)e * Tn + m];
          atomicAdd(&out[(size_t)tok * Dn + n], w * (acc[mt][nt][r] + bias));
        }
      }
    }
  }
}

// ---------------------------------------------------------------------------
// Host-side orchestration (graph-capture safe: only async memset + launches).
// Workspace layout (needs ~134.5 MB):
//   [0,32)        expert counters (8 int)
//   [64,100)      exclusive offsets (9 int)
//   [256, +E*T*4) token index lists
//   [.., +E*T*4)  routing weight lists
//   [262400, +2*T*F*4) GELU intermediate h (exactly 2T compacted rows)
// ---------------------------------------------------------------------------
extern "C" void kernel_launch(void* const* d_in, const int* in_sizes, int n_in,
                              void* d_out, int out_size, void* d_ws,
                              size_t ws_size, hipStream_t stream) {
  (void)in_sizes; (void)n_in; (void)out_size; (void)ws_size;
  const float* x     = (const float*)d_in[0];
  const float* noise = (const float*)d_in[1];
  const float* Wg    = (const float*)d_in[2];
  const float* bg    = (const float*)d_in[3];
  const float* Wn    = (const float*)d_in[4];
  const float* bn    = (const float*)d_in[5];
  const float* W1    = (const float*)d_in[6];
  const float* b1    = (const float*)d_in[7];
  const float* W2    = (const float*)d_in[8];
  const float* b2    = (const float*)d_in[9];

  float* out       = (float*)d_out;                    // [T, D]
  float* noisy_out = out + (size_t)Tn * Dn;            // [T, E]
  float* gate_out  = noisy_out + (size_t)Tn * En;      // [T, E]

  char* ws = (char*)d_ws;
  int*   cnt  = (int*)ws;
  int*   offs = (int*)(ws + 64);
  int*   idx  = (int*)(ws + 256);
  float* cw   = (float*)(ws + 256 + (size_t)En * Tn * 4);
  float* h    = (float*)(ws + 256 + (size_t)2 * En * Tn * 4);  // 262400

  (void)hipMemsetAsync(cnt, 0, 64, stream);
  (void)hipMemsetAsync(out, 0, (size_t)Tn * Dn * sizeof(float), stream);

  moe_router<<<Tn / 8, 256, 0, stream>>>(x, noise, Wg, bg, Wn, bn, noisy_out,
                                         gate_out, cnt, idx, cw);
  moe_offsets<<<1, 32, 0, stream>>>(cnt, offs);

  dim3 g1(Tn / BM, Fn / BN, En);  // 64 x 32 x 8, inactive tiles exit early
  moe_fc1<<<g1, 256, 0, stream>>>(x, W1, b1, cnt, offs, idx, h);

  dim3 g2(Tn / BM, Dn / BN, En);  // 64 x 8 x 8
  moe_fc2<<<g2, 256, 0, stream>>>(h, W2, b2, cnt, offs, idx, cw, out);
}